// Gat_31138512896563
// MI455X (gfx1250) — compile-verified
//
#include <hip/hip_runtime.h>
#include <math.h>

#define N_NODES 10000
#define N_EDGES 160000
#define D_FEAT  512
#define EN_TOT  (N_EDGES + N_NODES)

typedef float v2f __attribute__((ext_vector_type(2)));
typedef float v8f __attribute__((ext_vector_type(8)));

// ---------------- float <-> ordered uint key (for atomic max of signed floats) ---
__device__ __forceinline__ unsigned f2key(float f) {
  unsigned u = __float_as_uint(f);
  return (u & 0x80000000u) ? ~u : (u | 0x80000000u);
}
__device__ __forceinline__ float key2f(unsigned k) {
  return (k & 0x80000000u) ? __uint_as_float(k & 0x7FFFFFFFu) : __uint_as_float(~k);
}

// ---------------- 0: zero output accumulator + per-node reductions ---------------
__global__ void init_kernel(float* __restrict__ out,
                            unsigned* __restrict__ maxkey,
                            float* __restrict__ expsum) {
  int i = blockIdx.x * blockDim.x + threadIdx.x;
  if (i < N_NODES * D_FEAT) out[i] = 0.0f;
  if (i < N_NODES) { maxkey[i] = 0u; expsum[i] = 0.0f; }
}

// ---------------- 1: h = x @ W^T via V_WMMA_F32_16X16X4_F32 ----------------------
// Wave tile: 16 rows x 64 cols (4 accumulators). K loop step 4.
// A 16x4 layout: lane(0..15)=M, VGPR0/1 = K {2*hi, 2*hi+1}  (hi = lane>=16)
// B 4x16 layout: lane%16 = N, VGPR0/1 = K {2*hi, 2*hi+1}; B col n = W row (col_base+n)
__global__ __launch_bounds__(256) void gemm_wmma_kernel(const float* __restrict__ x,
                                                        const float* __restrict__ W,
                                                        float* __restrict__ h) {
  const int lane = threadIdx.x & 31;
  const int wave = threadIdx.x >> 5;          // 0..7 -> column chunk of 64
  const int m    = lane & 15;
  const int hi   = lane >> 4;                 // 0 or 1
  const int row_base = blockIdx.x * 16;       // 625 * 16 == 10000 exactly
  const int col_base = wave * 64;

  const float* ap = x + (size_t)(row_base + m) * D_FEAT + 2 * hi;
  const float* b0 = W + (size_t)(col_base +  0 + m) * D_FEAT + 2 * hi;
  const float* b1 = W + (size_t)(col_base + 16 + m) * D_FEAT + 2 * hi;
  const float* b2 = W + (size_t)(col_base + 32 + m) * D_FEAT + 2 * hi;
  const float* b3 = W + (size_t)(col_base + 48 + m) * D_FEAT + 2 * hi;

  v8f acc0 = {}, acc1 = {}, acc2 = {}, acc3 = {};

  #pragma unroll 4
  for (int k = 0; k < D_FEAT; k += 4) {
    v2f a  = *(const v2f*)(ap + k);
    v2f v0 = *(const v2f*)(b0 + k);
    v2f v1 = *(const v2f*)(b1 + k);
    v2f v2 = *(const v2f*)(b2 + k);
    v2f v3 = *(const v2f*)(b3 + k);
    acc0 = __builtin_amdgcn_wmma_f32_16x16x4_f32(false, a, false, v0, (short)0, acc0, false, false);
    acc1 = __builtin_amdgcn_wmma_f32_16x16x4_f32(false, a, false, v1, (short)0, acc1, false, false);
    acc2 = __builtin_amdgcn_wmma_f32_16x16x4_f32(false, a, false, v2, (short)0, acc2, false, false);
    acc3 = __builtin_amdgcn_wmma_f32_16x16x4_f32(false, a, false, v3, (short)0, acc3, false, false);
  }

  // C/D layout: VGPR v -> row M = v + 8*hi, col N = lane&15
  #pragma unroll
  for (int v = 0; v < 8; ++v) {
    float* hr = h + (size_t)(row_base + v + 8 * hi) * D_FEAT + col_base + m;
    hr[0]  = acc0[v];
    hr[16] = acc1[v];
    hr[32] = acc2[v];
    hr[48] = acc3[v];
  }
}

// ---------------- 2: e_src[n] = h[n].a_src ; e_dst[n] = h[n].a_dst ---------------
__global__ __launch_bounds__(256) void dots_kernel(const float* __restrict__ h,
                                                   const float* __restrict__ a_src,
                                                   const float* __restrict__ a_dst,
                                                   float* __restrict__ e_src,
                                                   float* __restrict__ e_dst) {
  int node = (int)((blockIdx.x * blockDim.x + threadIdx.x) >> 5);
  int lane = threadIdx.x & 31;
  if (node >= N_NODES) return;
  const float* hp = h + (size_t)node * D_FEAT;
  float s1 = 0.0f, s2 = 0.0f;
  #pragma unroll
  for (int t = 0; t < D_FEAT; t += 32) {
    float v = hp[t + lane];
    s1 += v * a_src[t + lane];
    s2 += v * a_dst[t + lane];
  }
  #pragma unroll
  for (int off = 16; off > 0; off >>= 1) {
    s1 += __shfl_xor(s1, off, 32);
    s2 += __shfl_xor(s2, off, 32);
  }
  if (lane == 0) { e_src[node] = s1; e_dst[node] = s2; }
}

// ---------------- 3: per-edge logits + segment max (self-loops appended) ---------
__global__ void edge_logits_kernel(const int* __restrict__ ei,
                                   const float* __restrict__ e_src,
                                   const float* __restrict__ e_dst,
                                   float* __restrict__ e_val,
                                   unsigned* __restrict__ maxkey) {
  int i = blockIdx.x * blockDim.x + threadIdx.x;
  if (i >= EN_TOT) return;
  int s, d;
  if (i < N_EDGES) { s = ei[i]; d = ei[N_EDGES + i]; }
  else             { s = d = i - N_EDGES; }
  float e = e_src[s] + e_dst[d];
  e = (e > 0.0f) ? e : 0.2f * e;             // leaky_relu(0.2)
  e_val[i] = e;
  atomicMax(&maxkey[d], f2key(e));
}

// ---------------- 4: exp(e - max[d]) + segment sum -------------------------------
__global__ void edge_exp_kernel(const int* __restrict__ ei,
                                float* __restrict__ e_val,
                                const unsigned* __restrict__ maxkey,
                                float* __restrict__ expsum) {
  int i = blockIdx.x * blockDim.x + threadIdx.x;
  if (i >= EN_TOT) return;
  int d = (i < N_EDGES) ? ei[N_EDGES + i] : (i - N_EDGES);
  float ex = expf(e_val[i] - key2f(maxkey[d]));
  e_val[i] = ex;
  atomicAdd(&expsum[d], ex);
}

// ---------------- 5: out[d] += alpha * h[s]  (one wave per edge) -----------------
__global__ __launch_bounds__(256) void edge_scatter_kernel(const int* __restrict__ ei,
                                                           const float* __restrict__ e_val,
                                                           const float* __restrict__ expsum,
                                                           const float* __restrict__ h,
                                                           float* __restrict__ agg) {
  int wid  = (int)((blockIdx.x * blockDim.x + threadIdx.x) >> 5);
  int lane = threadIdx.x & 31;
  if (wid >= EN_TOT) return;
  int s, d;
  if (wid < N_EDGES) { s = ei[wid]; d = ei[N_EDGES + wid]; }
  else               { s = d = wid - N_EDGES; }
  float alpha = e_val[wid] / expsum[d];
  const float* hs = h   + (size_t)s * D_FEAT;
  float*       od = agg + (size_t)d * D_FEAT;
  #pragma unroll 4
  for (int t = lane; t < D_FEAT; t += 32) {
    atomicAdd(&od[t], alpha * hs[t]);        // L2-resident float atomics
  }
}

// ---------------- 6: out = x + elu(agg + bias) -----------------------------------
__global__ void finalize_kernel(const float* __restrict__ x,
                                const float* __restrict__ bias,
                                float* __restrict__ out) {
  int i = blockIdx.x * blockDim.x + threadIdx.x;
  if (i >= N_NODES * D_FEAT) return;
  int col = i & (D_FEAT - 1);
  float z  = out[i] + bias[col];
  float el = (z > 0.0f) ? z : (expf(z) - 1.0f);
  out[i] = x[i] + el;
}

extern "C" void kernel_launch(void* const* d_in, const int* in_sizes, int n_in,
                              void* d_out, int out_size, void* d_ws, size_t ws_size,
                              hipStream_t stream) {
  (void)in_sizes; (void)n_in; (void)out_size; (void)ws_size;

  const float* x     = (const float*)d_in[0];   // [N, D]
  const int*   ei    = (const int*)  d_in[1];   // [2, E] flat: src then dst
  const float* W     = (const float*)d_in[2];   // [D, D]
  const float* a_src = (const float*)d_in[3];   // [D]
  const float* a_dst = (const float*)d_in[4];   // [D]
  const float* bias  = (const float*)d_in[5];   // [D]
  float* out = (float*)d_out;                   // [N, D] (also used as accumulator)

  // workspace layout
  float*    h      = (float*)d_ws;                      // N*D
  float*    e_src  = h + (size_t)N_NODES * D_FEAT;      // N
  float*    e_dst  = e_src + N_NODES;                   // N
  float*    expsum = e_dst + N_NODES;                   // N
  unsigned* maxkey = (unsigned*)(expsum + N_NODES);     // N
  float*    e_val  = (float*)(maxkey + N_NODES);        // E+N

  const int elemBlocks = (N_NODES * D_FEAT + 255) / 256;     // 20000
  const int edgeBlocks = (EN_TOT + 255) / 256;               // 665

  init_kernel<<<elemBlocks, 256, 0, stream>>>(out, maxkey, expsum);

  gemm_wmma_kernel<<<N_NODES / 16, 256, 0, stream>>>(x, W, h);

  dots_kernel<<<(N_NODES + 7) / 8, 256, 0, stream>>>(h, a_src, a_dst, e_src, e_dst);

  edge_logits_kernel<<<edgeBlocks, 256, 0, stream>>>(ei, e_src, e_dst, e_val, maxkey);

  edge_exp_kernel<<<edgeBlocks, 256, 0, stream>>>(ei, e_val, maxkey, expsum);

  edge_scatter_kernel<<<(EN_TOT + 7) / 8, 256, 0, stream>>>(ei, e_val, expsum, h, out);

  finalize_kernel<<<elemBlocks, 256, 0, stream>>>(x, bias, out);
}